// MultiScaleGNN_54769422958784
// MI455X (gfx1250) — compile-verified
//
#include <hip/hip_runtime.h>
#include <math.h>

// MultiScaleGNN for MI455X (gfx1250, wave32).
// Dense GEMMs via V_WMMA_F32_16X16X4_F32 (fp32 matrix pipe); edge phases are
// L2-resident f32-atomic scatter/gather (out arrays = 25.6MB << 192MB L2).

#define N_NODESc 50000
#define N_EDGESc 800000
#define E_TOTc   (N_EDGESc + N_NODESc)
#define Dc 128
#define HEADSc 4
#define HDIMc 32
#define NEG_SLOPEc 0.2f

typedef float v2f __attribute__((ext_vector_type(2)));
typedef float v8f __attribute__((ext_vector_type(8)));

static __device__ __forceinline__ void edge_nodes(const int* __restrict__ srcs,
                                                  const int* __restrict__ dsts,
                                                  int e, int& s, int& d) {
  if (e < N_EDGESc) { s = srcs[e]; d = dsts[e]; }
  else              { s = e - N_EDGESc; d = s; }   // self loops
}

// float atomic max via signed/unsigned integer atomics (valid with -inf init)
static __device__ __forceinline__ void atomicMaxF(float* addr, float val) {
  if (val >= 0.0f) atomicMax((int*)addr, __float_as_int(val));
  else             atomicMin((unsigned int*)addr, __float_as_uint(val));
}

// ---------------- degree (computed once; depends only on dst) ---------------
__global__ __launch_bounds__(256) void deg_init(float* __restrict__ deg) {
  int t = blockIdx.x * 256 + threadIdx.x;
  if (t < N_NODESc) deg[t] = 1.0f;                 // self loop
}
__global__ __launch_bounds__(256) void deg_count(const int* __restrict__ dsts,
                                                 float* __restrict__ deg) {
  int t = blockIdx.x * 256 + threadIdx.x;
  if (t < N_EDGESc) atomicAdd(&deg[dsts[t]], 1.0f);
}
__global__ __launch_bounds__(256) void deg_fin(float* __restrict__ deg) {
  int t = blockIdx.x * 256 + threadIdx.x;
  if (t < N_NODESc) deg[t] = rsqrtf(deg[t]);
}

// ---------------- GEMM: C[N,128] = A[N,128] @ W[128,128], fp32 WMMA ---------
// One wave per 16x16 C tile; K unrolled 128/4 = 32 V_WMMA_F32_16X16X4_F32.
__global__ __launch_bounds__(256) void gemm_wmma(const float* __restrict__ A,
                                                 const float* __restrict__ W,
                                                 float* __restrict__ C) {
  const int wave = (blockIdx.x * 256 + threadIdx.x) >> 5;
  const int lane = threadIdx.x & 31;
  const int tm = wave >> 3;          // row tile (N/16 = 3125)
  const int tn = wave & 7;           // col tile (128/16 = 8)
  if (tm * 16 >= N_NODESc) return;   // uniform per wave: EXEC stays all-1
  const int half = lane >> 4;        // 0: K={k,k+1}   1: K={k+2,k+3}
  const int l    = lane & 15;
  const float* __restrict__ arow = A + (size_t)(tm * 16 + l) * Dc;
  const float* __restrict__ wcol = W + tn * 16 + l;
  v8f c = {};
#pragma unroll 4
  for (int k = 0; k < Dc; k += 4) {
    const int kk = k + 2 * half;
    v2f a; a.x = arow[kk];            a.y = arow[kk + 1];
    v2f b; b.x = wcol[(size_t)kk * Dc]; b.y = wcol[(size_t)(kk + 1) * Dc];
    c = __builtin_amdgcn_wmma_f32_16x16x4_f32(false, a, false, b,
                                              (short)0, c, false, false);
  }
  // D layout: VGPR r -> M = r + 8*half, N = l
  float* __restrict__ crow = C + (size_t)tn * 16 + l;
  const int mb = tm * 16 + 8 * half;
#pragma unroll
  for (int r = 0; r < 8; ++r) crow[(size_t)(mb + r) * Dc] = c[r];
}

// ---------------- row init with broadcast bias ------------------------------
__global__ __launch_bounds__(256) void row_bias_init(float* __restrict__ out,
                                                     const float* __restrict__ bias) {
  int t = blockIdx.x * 256 + threadIdx.x;
  if (t < N_NODESc * Dc) out[t] = bias[t & (Dc - 1)];
}

// ---------------- GCN scatter: out[dst] += xw[src] * dinv[s]*dinv[d] --------
__global__ __launch_bounds__(256) void gcn_scatter(const float* __restrict__ xw,
                                                   const int* __restrict__ srcs,
                                                   const int* __restrict__ dsts,
                                                   const float* __restrict__ dinv,
                                                   float* __restrict__ out) {
  int t = blockIdx.x * 256 + threadIdx.x;          // E_TOT*32 = 27.2M < 2^31
  int e = t >> 5;
  if (e >= E_TOTc) return;
  int lane = t & 31;
  int s, d; edge_nodes(srcs, dsts, e, s, d);
  float nrm = dinv[s] * dinv[d];
  const float4 v = *(const float4*)(xw + (size_t)s * Dc + lane * 4);
  float* o = out + (size_t)d * Dc + lane * 4;
  atomicAdd(o + 0, v.x * nrm);
  atomicAdd(o + 1, v.y * nrm);
  atomicAdd(o + 2, v.z * nrm);
  atomicAdd(o + 3, v.w * nrm);
}

// ---------------- GAT per-node attention coefficients -----------------------
__global__ __launch_bounds__(256) void gat_node(const float* __restrict__ xw,
                                                const float* __restrict__ asv,
                                                const float* __restrict__ adv,
                                                float* __restrict__ a_src,
                                                float* __restrict__ a_dst) {
  int t = blockIdx.x * 256 + threadIdx.x;          // (node, head)
  if (t >= N_NODESc * HEADSc) return;
  int h = t & (HEADSc - 1);
  int node = t >> 2;
  const float* v = xw + (size_t)node * Dc + h * HDIMc;
  float ss = 0.f, sd = 0.f;
#pragma unroll
  for (int j = 0; j < HDIMc; ++j) {
    ss = fmaf(v[j], asv[h * HDIMc + j], ss);
    sd = fmaf(v[j], adv[h * HDIMc + j], sd);
  }
  a_src[t] = ss;
  a_dst[t] = sd;
}

__global__ __launch_bounds__(256) void init_mz(float* __restrict__ m,
                                               float* __restrict__ z) {
  int t = blockIdx.x * 256 + threadIdx.x;
  if (t < N_NODESc * HEADSc) { m[t] = __int_as_float(0xFF800000); z[t] = 0.f; }
}

__global__ __launch_bounds__(256) void gat_edge_max(const int* __restrict__ srcs,
                                                    const int* __restrict__ dsts,
                                                    const float* __restrict__ a_src,
                                                    const float* __restrict__ a_dst,
                                                    float* __restrict__ e_buf,
                                                    float* __restrict__ m) {
  int t = blockIdx.x * 256 + threadIdx.x;          // (edge, head), 3.4M
  if (t >= E_TOTc * HEADSc) return;
  int e = t >> 2, h = t & 3;
  int s, d; edge_nodes(srcs, dsts, e, s, d);
  float ev = a_src[s * HEADSc + h] + a_dst[d * HEADSc + h];
  ev = (ev > 0.f) ? ev : ev * NEG_SLOPEc;          // leaky relu
  e_buf[t] = ev;
  atomicMaxF(&m[d * HEADSc + h], ev);
}

__global__ __launch_bounds__(256) void gat_edge_sum(const int* __restrict__ srcs,
                                                    const int* __restrict__ dsts,
                                                    float* __restrict__ e_buf,
                                                    const float* __restrict__ m,
                                                    float* __restrict__ z) {
  int t = blockIdx.x * 256 + threadIdx.x;
  if (t >= E_TOTc * HEADSc) return;
  int e = t >> 2, h = t & 3;
  int s, d; edge_nodes(srcs, dsts, e, s, d);
  (void)s;
  float ee = __expf(e_buf[t] - m[d * HEADSc + h]);
  e_buf[t] = ee;                                   // keep exp'd logit
  atomicAdd(&z[d * HEADSc + h], ee);
}

// ---------------- GAT weighted scatter: out[dst] += xw[src] * alpha ---------
__global__ __launch_bounds__(256) void gat_scatter(const float* __restrict__ xw,
                                                   const int* __restrict__ srcs,
                                                   const int* __restrict__ dsts,
                                                   const float* __restrict__ ee,
                                                   const float* __restrict__ z,
                                                   float* __restrict__ out) {
  int t = blockIdx.x * 256 + threadIdx.x;
  int e = t >> 5;
  if (e >= E_TOTc) return;
  int lane = t & 31;
  int s, d; edge_nodes(srcs, dsts, e, s, d);
  int h = lane >> 3;                               // 8 lanes (32 chans) per head
  float alpha = ee[e * HEADSc + h] / z[d * HEADSc + h];
  const float4 v = *(const float4*)(xw + (size_t)s * Dc + lane * 4);
  float* o = out + (size_t)d * Dc + lane * 4;
  atomicAdd(o + 0, v.x * alpha);
  atomicAdd(o + 1, v.y * alpha);
  atomicAdd(o + 2, v.z * alpha);
  atomicAdd(o + 3, v.w * alpha);
}

// ---------------- residual + layernorm + relu (one wave per node) -----------
__global__ __launch_bounds__(256) void finalize_ln(const float* __restrict__ acc,
                                                   const float* __restrict__ hg,
                                                   const float* __restrict__ g,
                                                   const float* __restrict__ b,
                                                   float* __restrict__ out) {
  int t = blockIdx.x * 256 + threadIdx.x;
  int node = t >> 5;
  if (node >= N_NODESc) return;
  int lane = t & 31;
  size_t base = (size_t)node * Dc + lane * 4;
  float4 a = *(const float4*)(acc + base);
  float4 h = *(const float4*)(hg + base);
  float v0 = a.x + h.x, v1 = a.y + h.y, v2 = a.z + h.z, v3 = a.w + h.w;
  float s = v0 + v1 + v2 + v3;
#pragma unroll
  for (int o = 16; o > 0; o >>= 1) s += __shfl_xor(s, o, 32);
  float mu = s * (1.0f / Dc);
  float d0 = v0 - mu, d1 = v1 - mu, d2 = v2 - mu, d3 = v3 - mu;
  float q = d0 * d0 + d1 * d1 + d2 * d2 + d3 * d3;
#pragma unroll
  for (int o = 16; o > 0; o >>= 1) q += __shfl_xor(q, o, 32);
  float ri = rsqrtf(q * (1.0f / Dc) + 1e-5f);
  const float4 gg = *(const float4*)(g + lane * 4);
  const float4 bb = *(const float4*)(b + lane * 4);
  float4 r;
  r.x = fmaxf(fmaf(d0 * ri, gg.x, bb.x), 0.0f);
  r.y = fmaxf(fmaf(d1 * ri, gg.y, bb.y), 0.0f);
  r.z = fmaxf(fmaf(d2 * ri, gg.z, bb.z), 0.0f);
  r.w = fmaxf(fmaf(d3 * ri, gg.w, bb.w), 0.0f);
  *(float4*)(out + base) = r;
}

extern "C" void kernel_launch(void* const* d_in, const int* in_sizes, int n_in,
                              void* d_out, int out_size, void* d_ws, size_t ws_size,
                              hipStream_t stream) {
  (void)in_sizes; (void)n_in; (void)out_size; (void)ws_size;
  const float* x       = (const float*)d_in[0];
  const int*   ei      = (const int*)d_in[1];
  const float* gcn_w   = (const float*)d_in[2];
  const float* gcn_b   = (const float*)d_in[3];
  const float* gat_w   = (const float*)d_in[4];
  const float* att_src = (const float*)d_in[5];
  const float* att_dst = (const float*)d_in[6];
  const float* gat_b   = (const float*)d_in[7];
  const float* ln_g    = (const float*)d_in[8];
  const float* ln_b    = (const float*)d_in[9];
  const int* srcs = ei;
  const int* dsts = ei + N_EDGESc;

  // workspace layout (floats): 4 N*D buffers + small arrays (~120 MB total)
  float* ws = (float*)d_ws;
  const size_t ND = (size_t)N_NODESc * Dc;
  float* bufA  = ws;            // h (layer input / output ping)
  float* bufB  = bufA + ND;     // GCN xw, recycled as GAT accumulator
  float* bufC  = bufB + ND;     // h_gcn
  float* bufD  = bufC + ND;     // GAT xw
  float* dinv  = bufD + ND;                           // N
  float* aS    = dinv + N_NODESc;                     // N*H
  float* aD    = aS + (size_t)N_NODESc * HEADSc;      // N*H
  float* mbuf  = aD + (size_t)N_NODESc * HEADSc;      // N*H
  float* zbuf  = mbuf + (size_t)N_NODESc * HEADSc;    // N*H
  float* e_buf = zbuf + (size_t)N_NODESc * HEADSc;    // E_TOT*H

  hipMemcpyAsync(bufA, x, ND * sizeof(float), hipMemcpyDeviceToDevice, stream);

  const int BN   = (N_NODESc + 255) / 256;
  const int BE   = (N_EDGESc + 255) / 256;
  const int BND  = (N_NODESc * Dc + 255) / 256;
  const int BNH  = (N_NODESc * HEADSc + 255) / 256;
  const int BEH  = (E_TOTc * HEADSc + 255) / 256;
  const int BE32 = (int)(((long long)E_TOTc * 32 + 255) / 256);
  const int BN32 = (N_NODESc * 32 + 255) / 256;
  const int BGEMM = (N_NODESc / 16) * (Dc / 16) / 8;  // 8 waves per block

  deg_init<<<BN, 256, 0, stream>>>(dinv);
  deg_count<<<BE, 256, 0, stream>>>(dsts, dinv);
  deg_fin<<<BN, 256, 0, stream>>>(dinv);

  for (int layer = 0; layer < 3; ++layer) {
    // ---- GCN conv ----
    gemm_wmma<<<BGEMM, 256, 0, stream>>>(bufA, gcn_w + (size_t)layer * Dc * Dc, bufB);
    row_bias_init<<<BND, 256, 0, stream>>>(bufC, gcn_b + layer * Dc);
    gcn_scatter<<<BE32, 256, 0, stream>>>(bufB, srcs, dsts, dinv, bufC);
    // ---- GAT conv on h_gcn ----
    gemm_wmma<<<BGEMM, 256, 0, stream>>>(bufC, gat_w + (size_t)layer * Dc * Dc, bufD);
    gat_node<<<BNH, 256, 0, stream>>>(bufD,
                                      att_src + layer * HEADSc * HDIMc,
                                      att_dst + layer * HEADSc * HDIMc, aS, aD);
    init_mz<<<BNH, 256, 0, stream>>>(mbuf, zbuf);
    gat_edge_max<<<BEH, 256, 0, stream>>>(srcs, dsts, aS, aD, e_buf, mbuf);
    gat_edge_sum<<<BEH, 256, 0, stream>>>(srcs, dsts, e_buf, mbuf, zbuf);
    row_bias_init<<<BND, 256, 0, stream>>>(bufB, gat_b + layer * Dc);
    gat_scatter<<<BE32, 256, 0, stream>>>(bufD, srcs, dsts, e_buf, zbuf, bufB);
    // ---- residual + layernorm + relu ----
    float* outp = (layer == 2) ? (float*)d_out : bufA;
    finalize_ln<<<BN32, 256, 0, stream>>>(bufB, bufC, ln_g + layer * Dc,
                                          ln_b + layer * Dc, outp);
  }
}